// EphapticCoupling_51857435132573
// MI455X (gfx1250) — compile-verified
//
#include <hip/hip_runtime.h>

// EphapticCoupling for MI455X (gfx1250, wave32).
//
// Roofline: must stream x (128 MiB) in and out (128 MiB) -> ~11 us floor at
// 23.3 TB/s HBM. One workgroup per batch: phase 1 reads the 256 KB batch slice
// for row sums (S), phase 2 does the 32x32 @ 32x3 decay matmul with
// v_wmma_f32_16x16x4_f32, phase 3 re-reads x (L2 hit: whole x fits in 192 MB
// L2 and reuse is immediate per-WG) and writes out. w (24 KB) is staged into
// LDS with async global->LDS loads overlapped with phase 1.

#define BATCH 512
#define MODS  32
#define DIM   2048
#define CSCALE (0.1f / 2048.0f)

typedef float v2f __attribute__((ext_vector_type(2)));
typedef float v8f __attribute__((ext_vector_type(8)));

// Exact pointee type expected by __builtin_amdgcn_global_load_async_to_lds_b128
// per the compiler diagnostic: int __attribute__((vector_size(16))).
typedef int eph_v4i __attribute__((vector_size(4 * sizeof(int))));
typedef __attribute__((address_space(1))) eph_v4i* eph_gptr;
typedef __attribute__((address_space(3))) eph_v4i* eph_lptr;

__device__ __forceinline__ float decay_val(int i, int j) {
    // exp(-|i-j|/2), zero diagonal
    return (i == j) ? 0.0f : __expf(-0.5f * fabsf((float)(i - j)));
}

__global__ __launch_bounds__(256)
void EphapticCoupling_kernel(const float* __restrict__ x,
                             const float* __restrict__ w,
                             float* __restrict__ out) {
    __shared__ float wl[DIM * 3];        // 24 KB: conv weights, row-major [d][k]
    __shared__ float S_lds[MODS * 16];   // 2 KB: S padded to 16 cols (cols 0..2 valid)
    __shared__ float SD_lds[MODS * 16];  // 2 KB: pre-scaled decay@S (cols 0..2 valid)

    const int b    = blockIdx.x;
    const int t    = threadIdx.x;
    const int wave = t >> 5;
    const int lane = t & 31;

    const float* xb = x   + (size_t)b * (MODS * DIM);
    float*       ob = out + (size_t)b * (MODS * DIM);

    // ---- stage w into LDS; async path overlaps with phase-1 reductions ----
#if defined(__gfx1250__) && __has_builtin(__builtin_amdgcn_global_load_async_to_lds_b128)
#define EPH_ASYNC_W 1
    {
        // 6144 floats = 1536 float4; each of 256 lanes issues 6 async b128 ops.
        for (int q = 0; q < 6; ++q) {
            const int f4 = t + q * 256;
            __builtin_amdgcn_global_load_async_to_lds_b128(
                (eph_gptr)(w + (size_t)f4 * 4),
                (eph_lptr)(&wl[f4 * 4]),
                0, 0);
        }
    }
#else
    for (int q = t; q < DIM * 3; q += 256) wl[q] = w[q];
#endif

    // ---- phase 1: row sums -> S[row] = {T - x_last, T, T - x_first} ----
    // 8 waves x 4 rows each; lane-strided float4 loads (contiguous per wave).
    for (int r = 0; r < 4; ++r) {
        const int row = wave * 4 + r;
        const float4* xr = (const float4*)(xb + (size_t)row * DIM);
        float sum = 0.0f;
#pragma unroll
        for (int it = 0; it < 16; ++it) {
            const float4 v = xr[lane + 32 * it];
            sum += (v.x + v.y) + (v.z + v.w);
        }
#pragma unroll
        for (int m = 16; m >= 1; m >>= 1) sum += __shfl_xor(sum, m, 32);
        if (lane == 0) {
            const float xf = xb[(size_t)row * DIM];
            const float xl = xb[(size_t)row * DIM + (DIM - 1)];
            S_lds[row * 16 + 0] = sum - xl;  // tap offset -1: drop last element
            S_lds[row * 16 + 1] = sum;       // tap offset  0
            S_lds[row * 16 + 2] = sum - xf;  // tap offset +1: drop first element
        }
    }

#ifdef EPH_ASYNC_W
#if __has_builtin(__builtin_amdgcn_s_wait_asynccnt)
    __builtin_amdgcn_s_wait_asynccnt(0);
#else
    asm volatile("s_wait_asynccnt 0" ::: "memory");
#endif
#endif
    __syncthreads();

    // ---- phase 2: SD = (0.1/D) * decay @ S via v_wmma_f32_16x16x4_f32 ----
    // Two 16x16 output tiles (waves 0 and 1), K=32 split into 8 steps of 4.
    // A (16x4 f32): lane m=lane&15, vgpr0 -> K = 4*ks + 2*half, vgpr1 -> +1.
    // B (4x16 f32): mirrored K-split across lane halves, n = lane&15.
    // Only SD columns 0..2 are consumed later, so B columns 3..15 need no init.
    if (wave < 2) {
        const int half = lane >> 4;
        const int n    = lane & 15;
        const int mrow = (wave << 4) + (lane & 15);
        v8f acc = {};
#pragma unroll
        for (int ks = 0; ks < 8; ++ks) {
            const int j0 = ks * 4 + 2 * half;
            v2f a, bm;
            a.x  = decay_val(mrow, j0);
            a.y  = decay_val(mrow, j0 + 1);
            bm.x = S_lds[j0 * 16 + n];
            bm.y = S_lds[(j0 + 1) * 16 + n];
            acc = __builtin_amdgcn_wmma_f32_16x16x4_f32(false, a, false, bm,
                                                        (short)0, acc, false, false);
        }
        // C/D layout: vgpr r, lane-half h -> row (r + 8h), col n.
#pragma unroll
        for (int r = 0; r < 8; ++r)
            SD_lds[((wave << 4) + r + 8 * half) * 16 + n] = CSCALE * acc[r];
    }
    __syncthreads();

    // ---- phase 3: out = x + sum_k w[:,k] * SD[i,k]  (streaming, L2-hot x) ----
    // Thread t owns d = 4t..4t+3 and d = 4t+1024..4t+1027 so each wave's
    // float4 loads/stores are fully contiguous 512B bursts.
    const int dA = 4 * t;
    float4 cA0, cA1, cA2, cB0, cB1, cB2;
    {
        const float* p = &wl[dA * 3];
        cA0 = make_float4(p[0], p[3], p[6], p[9]);
        cA1 = make_float4(p[1], p[4], p[7], p[10]);
        cA2 = make_float4(p[2], p[5], p[8], p[11]);
        const float* q = &wl[(dA + 1024) * 3];
        cB0 = make_float4(q[0], q[3], q[6], q[9]);
        cB1 = make_float4(q[1], q[4], q[7], q[10]);
        cB2 = make_float4(q[2], q[5], q[8], q[11]);
    }
#pragma unroll 2
    for (int i = 0; i < MODS; ++i) {
        const float s0 = SD_lds[i * 16 + 0];
        const float s1 = SD_lds[i * 16 + 1];
        const float s2 = SD_lds[i * 16 + 2];
        const float4* xr   = (const float4*)(xb + (size_t)i * DIM);
        float4*       orow = (float4*)(ob + (size_t)i * DIM);
        float4 va = xr[t];
        float4 vb = xr[t + 256];
        va.x = fmaf(cA0.x, s0, fmaf(cA1.x, s1, fmaf(cA2.x, s2, va.x)));
        va.y = fmaf(cA0.y, s0, fmaf(cA1.y, s1, fmaf(cA2.y, s2, va.y)));
        va.z = fmaf(cA0.z, s0, fmaf(cA1.z, s1, fmaf(cA2.z, s2, va.z)));
        va.w = fmaf(cA0.w, s0, fmaf(cA1.w, s1, fmaf(cA2.w, s2, va.w)));
        vb.x = fmaf(cB0.x, s0, fmaf(cB1.x, s1, fmaf(cB2.x, s2, vb.x)));
        vb.y = fmaf(cB0.y, s0, fmaf(cB1.y, s1, fmaf(cB2.y, s2, vb.y)));
        vb.z = fmaf(cB0.z, s0, fmaf(cB1.z, s1, fmaf(cB2.z, s2, vb.z)));
        vb.w = fmaf(cB0.w, s0, fmaf(cB1.w, s1, fmaf(cB2.w, s2, vb.w)));
        orow[t]       = va;
        orow[t + 256] = vb;
    }
}

extern "C" void kernel_launch(void* const* d_in, const int* in_sizes, int n_in,
                              void* d_out, int out_size, void* d_ws, size_t ws_size,
                              hipStream_t stream) {
    (void)in_sizes; (void)n_in; (void)out_size; (void)d_ws; (void)ws_size;
    const float* x = (const float*)d_in[0];  // [512, 32, 2048] f32
    const float* w = (const float*)d_in[1];  // [2048, 3] f32
    float* out = (float*)d_out;              // [512, 32, 2048] f32
    EphapticCoupling_kernel<<<dim3(BATCH), dim3(256), 0, stream>>>(x, w, out);
}